// AdamCountSketch_53171695124624
// MI455X (gfx1250) — compile-verified
//
#include <hip/hip_runtime.h>
#include <hip/hip_bf16.h>

// ---------------------------------------------------------------------------
// AdamCountSketch for MI455X (gfx1250).
// Memory-bound (~0.7 GB traffic, ~30us roofline @ 23.3 TB/s). No GEMM -> no
// WMMA. CDNA5-specific wins used instead:
//   * 320KB LDS/WGP holds the ENTIRE 256KB CountSketch per workgroup:
//       - build pass: ds_add_f32 LDS atomics for the 16M scatter-adds,
//         then one 64K-entry global atomic merge per block.
//       - update pass: sketch async-staged to LDS once per block; all 16M
//         decompress gathers become ds_load_b32 (no random VMEM).
//   * GLOBAL_LOAD_ASYNC_TO_LDS_B128 staging (ASYNCcnt + s_wait_asynccnt),
//     double-buffered in the build pass to overlap streaming with scatter.
// ---------------------------------------------------------------------------

#define D_TOTAL   16777216
#define M_SKETCH  65536

// ---- sketch build pass ----
#define NT_SK     256                     // 8 wave32 waves per block
#define NBLK_SK   128                     // 1 block per WGP (256KB LDS each)
#define CHUNK     (D_TOTAL / NBLK_SK)     // 131072 elements per block
#define TILE      2048                    // elements per staged tile
#define TILES_PB  (CHUNK / TILE)          // 64 tiles per block
#define STAGE_W   (3 * TILE)              // grad + s + h words per buffer
#define LDS_SK_BYTES ((M_SKETCH + 2 * STAGE_W) * 4)   // 311296 B <= 320KB

// ---- update pass ----
#define NT_UP     1024                    // 32 wave32 waves (max) per WGP
#define NBLK_UP   128
#define N4_BLK    ((D_TOTAL / 4) / NBLK_UP)            // 32768 float4 / block
#define LDS_UP_BYTES (M_SKETCH * 4)                    // 256 KB

// 16-byte int vector matching the async builtin's expected pointee type
// (confirmed by round-1 diagnostic: v4i AS(1)* src, v4i AS(3)* dst, imm, imm).
typedef int v4i __attribute__((__vector_size__(16)));
typedef __attribute__((address_space(1))) v4i* global_v4i_ptr;
typedef __attribute__((address_space(3))) v4i* lds_v4i_ptr;

#if __has_builtin(__builtin_amdgcn_global_load_async_to_lds_b128)
  #define HAVE_ASYNC 1
  #if __has_builtin(__builtin_amdgcn_s_wait_asynccnt)
    #define WAIT_ASYNC(n) __builtin_amdgcn_s_wait_asynccnt(n)
  #else
    #define WAIT_ASYNC(n) asm volatile("s_wait_asynccnt %0" :: "n"(n) : "memory")
  #endif
#else
  #define HAVE_ASYNC 0
  #define WAIT_ASYNC(n) do {} while (0)
#endif

__device__ __forceinline__ void async_copy_b128(const void* gsrc, void* ldst) {
#if HAVE_ASYNC
  __builtin_amdgcn_global_load_async_to_lds_b128(
      (global_v4i_ptr)(gsrc), (lds_v4i_ptr)(ldst), /*offset=*/0, /*cpol=*/0);
#else
  *(float4*)ldst = *(const float4*)gsrc;
#endif
}

__device__ __forceinline__ void atomic_add_f32_global(float* p, float v) {
#if defined(__HIP_DEVICE_COMPILE__)
  // HIP default atomic semantics (no fine-grained/remote) -> native
  // global_atomic_add_f32 (no-return form since result is unused).
  __hip_atomic_fetch_add(p, v, __ATOMIC_RELAXED, __HIP_MEMORY_SCOPE_AGENT);
#else
  atomicAdd(p, v);
#endif
}

// ---------------------------------------------------------------------------
// Kernel 1: zero the global sketch (d_ws is poisoned / carries prior state).
// ---------------------------------------------------------------------------
__global__ void zero_sketch(float4* __restrict__ sk4) {
  int i = blockIdx.x * blockDim.x + threadIdx.x;   // 64 blk * 256 thr = 16384
  sk4[i] = make_float4(0.f, 0.f, 0.f, 0.f);
}

// ---------------------------------------------------------------------------
// Kernel 2: sketch[j] = sum_{h[i]==j} s[i]*grad[i], via LDS-resident partial
// sketches + double-buffered async staging.
// ---------------------------------------------------------------------------
__global__ __launch_bounds__(NT_SK)
void sketch_build(const float* __restrict__ grad,
                  const float* __restrict__ s,
                  const int*   __restrict__ h,
                  float*       __restrict__ gsk) {
  extern __shared__ float lds[];
  float* lsk   = lds;              // [M_SKETCH] local sketch (256 KB)
  float* stage = lds + M_SKETCH;   // 2 x STAGE_W staging (48 KB)
  const int tid  = threadIdx.x;
  const int base = blockIdx.x * CHUNK;

  for (int k = tid; k < M_SKETCH; k += NT_SK) lsk[k] = 0.0f;

  auto issue = [&](int t, int b) {         // 6 async b128 loads per thread
    const int g0 = base + t * TILE;
    float* sg = stage + b * STAGE_W;
    float* ss = sg + TILE;
    int*   sh = (int*)(ss + TILE);
    #pragma unroll
    for (int k = 0; k < TILE / 4 / NT_SK; ++k) {   // 2 float4 per array
      const int j = tid + k * NT_SK;
      async_copy_b128(((const float4*)(grad + g0)) + j, ((float4*)sg) + j);
      async_copy_b128(((const float4*)(s    + g0)) + j, ((float4*)ss) + j);
      async_copy_b128(((const int4*)  (h    + g0)) + j, ((int4*)  sh) + j);
    }
  };

  issue(0, 0);
  __syncthreads();                     // lsk zeroing visible

  for (int t = 0; t < TILES_PB; ++t) {
    const int b = t & 1;
    if (t + 1 < TILES_PB) {
      issue(t + 1, b ^ 1);             // prefetch next tile (6 more in flight)
      WAIT_ASYNC(6);                   // tile t's 6 loads have landed in LDS
    } else {
      WAIT_ASYNC(0);
    }
    __syncthreads();                   // tile t data visible to all waves

    float* sg = stage + b * STAGE_W;
    float* ss = sg + TILE;
    int*   sh = (int*)(ss + TILE);
    #pragma unroll
    for (int k = 0; k < TILE / NT_SK; ++k) {       // 8 elements per thread
      const int e = tid + k * NT_SK;
      atomicAdd(&lsk[sh[e]], sg[e] * ss[e]);       // ds_add_f32
    }
    __syncthreads();                   // buffer b free for reissue at t+2
  }

  for (int k = tid; k < M_SKETCH; k += NT_SK)
    atomic_add_f32_global(&gsk[k], lsk[k]);        // global_atomic_add_f32
}

// ---------------------------------------------------------------------------
// Kernel 3: fused decompress + Adam update. The 256KB sketch is async-staged
// into LDS once per block; all gathers are ds_load_b32. Streams are float4
// coalesced. 1024 threads = 32 waves/WGP for latency hiding at 1 WG/WGP.
// ---------------------------------------------------------------------------
__global__ __launch_bounds__(NT_UP)
void adam_update(const float4* __restrict__ p4,
                 const float4* __restrict__ m4,
                 const float4* __restrict__ v4,
                 const int4*   __restrict__ h4,
                 const float4* __restrict__ s4,
                 const float*  __restrict__ gsk,
                 const int*    __restrict__ stepPtr,
                 float4* __restrict__ op4,
                 float4* __restrict__ om4,
                 float4* __restrict__ ov4) {
  extern __shared__ float lsk[];                   // [M_SKETCH] sketch copy
  const int tid = threadIdx.x;

  // Stage sketch: 16384 float4 / 1024 thr = 16 async b128 loads per thread.
  #pragma unroll 4
  for (int k = tid; k < M_SKETCH / 4; k += NT_UP)
    async_copy_b128(((const float4*)gsk) + k, ((float4*)lsk) + k);

  // Match reference's Python-double bias-correction constants (overlaps with
  // the async staging above).
  const int    step = *stepPtr;
  const double bc1  = 1.0 - pow(0.9,   (double)step);
  const double bc2  = 1.0 - pow(0.999, (double)step);
  const float  lr_bc1 = (float)(1e-3 / bc1);       // LR / bc1
  const float  sbc2   = (float)sqrt(bc2);          // np.float32(np.sqrt(bc2))

  WAIT_ASYNC(0);
  __syncthreads();                                 // sketch visible in LDS

  const int base4 = blockIdx.x * N4_BLK;
  #pragma unroll 2
  for (int k = 0; k < N4_BLK / NT_UP; ++k) {       // 32 coalesced iterations
    const int i = base4 + tid + k * NT_UP;
    const int4   hh = h4[i];
    const float4 ss = s4[i];
    float4 g;                                      // decompress via LDS
    g.x = ss.x * lsk[hh.x];  g.y = ss.y * lsk[hh.y];
    g.z = ss.z * lsk[hh.z];  g.w = ss.w * lsk[hh.w];

    float4 m = m4[i], v = v4[i], p = p4[i];

    m.x = 0.9f * m.x + 0.1f * g.x;   v.x = 0.999f * v.x + 0.001f * g.x * g.x;
    m.y = 0.9f * m.y + 0.1f * g.y;   v.y = 0.999f * v.y + 0.001f * g.y * g.y;
    m.z = 0.9f * m.z + 0.1f * g.z;   v.z = 0.999f * v.z + 0.001f * g.z * g.z;
    m.w = 0.9f * m.w + 0.1f * g.w;   v.w = 0.999f * v.w + 0.001f * g.w * g.w;

    p.x -= lr_bc1 * (m.x / (sqrtf(v.x) / sbc2 + 1e-8f));
    p.y -= lr_bc1 * (m.y / (sqrtf(v.y) / sbc2 + 1e-8f));
    p.z -= lr_bc1 * (m.z / (sqrtf(v.z) / sbc2 + 1e-8f));
    p.w -= lr_bc1 * (m.w / (sqrtf(v.w) / sbc2 + 1e-8f));

    op4[i] = p;  om4[i] = m;  ov4[i] = v;
  }
}

// ---------------------------------------------------------------------------
extern "C" void kernel_launch(void* const* d_in, const int* in_sizes, int n_in,
                              void* d_out, int out_size, void* d_ws, size_t ws_size,
                              hipStream_t stream) {
  const float* p    = (const float*)d_in[0];
  const float* grad = (const float*)d_in[1];
  const float* m    = (const float*)d_in[2];
  const float* v    = (const float*)d_in[3];
  const int*   h    = (const int*)d_in[4];
  const float* s    = (const float*)d_in[5];
  const int*   step = (const int*)d_in[6];

  float* sk = (float*)d_ws;                 // 256 KB sketch scratch
  float* op = (float*)d_out;
  float* om = op + D_TOTAL;
  float* ov = om + D_TOTAL;

  // Allow >64KB dynamic LDS (320KB/WGP on CDNA5). Benign under graph capture.
  (void)hipFuncSetAttribute((const void*)sketch_build,
                            hipFuncAttributeMaxDynamicSharedMemorySize,
                            LDS_SK_BYTES);
  (void)hipFuncSetAttribute((const void*)adam_update,
                            hipFuncAttributeMaxDynamicSharedMemorySize,
                            LDS_UP_BYTES);

  zero_sketch <<<M_SKETCH / 4 / 256, 256, 0, stream>>>((float4*)sk);
  sketch_build<<<NBLK_SK, NT_SK, LDS_SK_BYTES, stream>>>(grad, s, h, sk);
  adam_update <<<NBLK_UP, NT_UP, LDS_UP_BYTES, stream>>>(
      (const float4*)p, (const float4*)m, (const float4*)v,
      (const int4*)h, (const float4*)s, sk, step,
      (float4*)op, (float4*)om, (float4*)ov);
}